// ForwardTransformLayer_41901700940340
// MI455X (gfx1250) — compile-verified
//
#include <hip/hip_runtime.h>

// DWT forward (8-tap circular conv + stride-2 decimation), 4096 x 16384 f32.
// out[0 .. 4096*8192)        = details      (wavelet filter)
// out[4096*8192 .. 2x)       = approximation (scaling filter)
//
// Formulation: out_f[r, n] = sum_s x[r, s] * h_f[2n - s],  s = 2n-7 .. 2n (mod 16384)
// mapped to V_WMMA_F32_16X16X4_F32 tiles: D(16x16) = sum_c A_c(16x4) * B_c(4x16)
// with A = input rows, B = constant banded filter matrix.
// Staging uses the CDNA5 async global->LDS path (ASYNCcnt).

#define SIG_LEN   16384
#define OUT_LEN   8192
#define NROWS     4096

#define ROWS_PB   16          // rows per block
#define COLS_PB   512         // output columns per block (per filter)
#define LDS_COLS  1032        // 2*COLS_PB + 8 halo (circular)
#define LDS_PITCH 1036        // multiple of 4 (16B rows); %64 = 12 -> conflict-free A loads
#define FOFF      (ROWS_PB * LDS_PITCH)   // filter table offset in LDS

typedef __attribute__((ext_vector_type(2))) float v2f;
typedef __attribute__((ext_vector_type(8))) float v8f;
typedef __attribute__((ext_vector_type(4))) int   v4i;

#define AS_GLOBAL __attribute__((address_space(1)))
#define AS_LOCAL  __attribute__((address_space(3)))

__global__ __launch_bounds__(256)
void dwt_fwd_wmma(const float* __restrict__ x,
                  const float* __restrict__ h,
                  float* __restrict__ out)
{
    __shared__ float lds[FOFF + 16];

    const int tid = threadIdx.x;
    const int rt  = blockIdx.x >> 4;       // row tile   (0..255)
    const int cs  = blockIdx.x & 15;       // col strip  (0..15)
    const int r0  = rt * ROWS_PB;
    const int c0  = cs * COLS_PB;

    // ---- filter table in LDS: [0..7]=scaling, [8..15]=wavelet (QMF) ----
    if (tid < 16) {
        if (tid < 8) {
            lds[FOFF + tid] = h[tid];
        } else {
            int k = tid - 8;
            float g = h[7 - k];                  // flip
            lds[FOFF + tid] = (k & 1) ? -g : g;  // negate odd taps
        }
    }

    // ---- stage input strip into LDS via async B128 copies (circular wrap) ----
    // columns s in [2*c0 - 8, 2*c0 + 1023], 1032 floats/row, 16 rows
    const int base4 = (2 * c0 - 8 + SIG_LEN) >> 2;          // float4 index
    for (int k2 = tid; k2 < ROWS_PB * (LDS_COLS / 4); k2 += 256) {
        int row = k2 / (LDS_COLS / 4);
        int t4  = k2 - row * (LDS_COLS / 4);
        int g4  = (base4 + t4) & (SIG_LEN / 4 - 1);
        float* gp = const_cast<float*>(x + (size_t)(r0 + row) * SIG_LEN + (size_t)g4 * 4);
        float* lp = &lds[row * LDS_PITCH + t4 * 4];
        __builtin_amdgcn_global_load_async_to_lds_b128(
            (AS_GLOBAL v4i*)gp, (AS_LOCAL v4i*)lp, /*offset=*/0, /*cpol=*/0);
    }
    __builtin_amdgcn_s_wait_asynccnt(0);
    __syncthreads();

    const int lane = tid & 31;
    const int wave = tid >> 5;             // 8 waves, 64 output cols each
    const int half = lane >> 4;            // lane half (ISA VGPR layout)
    const int nm   = lane & 15;            // M row for A, N col for B/C/D
    const int j0   = half * 2;             // K pair held by this lane

    // ---- constant banded filter fragments: B_c[j][n] = h[2n + 8 - 4c - j] ----
    v2f Bd[10], Ba[10];
#pragma unroll
    for (int c = 0; c < 10; ++c) {
        int i0 = 2 * nm + 8 - 4 * c - j0;  // tap index for K = j0
        int i1 = i0 - 1;                   // tap index for K = j0 + 1
        v2f bd, ba;
        bd.x = (i0 >= 0 && i0 < 8) ? lds[FOFF + 8 + i0] : 0.0f;
        bd.y = (i1 >= 0 && i1 < 8) ? lds[FOFF + 8 + i1] : 0.0f;
        ba.x = (i0 >= 0 && i0 < 8) ? lds[FOFF + i0] : 0.0f;
        ba.y = (i1 >= 0 && i1 < 8) ? lds[FOFF + i1] : 0.0f;
        Bd[c] = bd;
        Ba[c] = ba;
    }

    float* __restrict__ outd = out;                                   // details
    float* __restrict__ outa = out + (size_t)NROWS * OUT_LEN;         // approximation

#pragma unroll
    for (int i = 0; i < 4; ++i) {
        const int n0 = c0 + wave * 64 + i * 16;     // output col tile base
        const int tt = 2 * (wave * 64 + i * 16);    // LDS col of s = 2*n0 - 8

        v8f accd = {0.f,0.f,0.f,0.f,0.f,0.f,0.f,0.f};
        v8f acca = {0.f,0.f,0.f,0.f,0.f,0.f,0.f,0.f};

#pragma unroll
        for (int c = 0; c < 10; ++c) {
            // A fragment: lane nm holds row nm, K pair (j0, j0+1) of chunk c
            v2f a = *reinterpret_cast<const v2f*>(&lds[nm * LDS_PITCH + tt + 4 * c + j0]);
            // D = A x B + C   (chained accumulation over K chunks; A shared by both filters)
            accd = __builtin_amdgcn_wmma_f32_16x16x4_f32(
                       false, a, false, Bd[c], (short)0, accd, false, false);
            acca = __builtin_amdgcn_wmma_f32_16x16x4_f32(
                       false, a, false, Ba[c], (short)0, acca, false, false);
        }

        // C/D layout: VGPR v -> row (v + 8*half), lane -> col nm
#pragma unroll
        for (int v = 0; v < 8; ++v) {
            size_t off = (size_t)(r0 + v + 8 * half) * OUT_LEN + (size_t)(n0 + nm);
            outd[off] = accd[v];
            outa[off] = acca[v];
        }
    }
}

extern "C" void kernel_launch(void* const* d_in, const int* in_sizes, int n_in,
                              void* d_out, int out_size, void* d_ws, size_t ws_size,
                              hipStream_t stream) {
    const float* x = (const float*)d_in[0];   // 4096 x 16384 f32
    const float* h = (const float*)d_in[1];   // 8 f32 (scaling filter)
    float* out = (float*)d_out;               // 2 x 4096 x 8192 f32

    // 256 row-tiles x 16 col-strips = 4096 blocks, 256 threads (8 waves)
    dwt_fwd_wmma<<<dim3(4096), dim3(256), 0, stream>>>(x, h, out);
}